// RIA_73383811220015
// MI455X (gfx1250) — compile-verified
//
#include <hip/hip_runtime.h>
#include <math.h>

typedef __attribute__((ext_vector_type(2))) float v2f;
typedef __attribute__((ext_vector_type(8))) float v8f;

#define IMG_H 300
#define IMG_W 300
#define BATCH 32
#define NTX   19            // ceil(300/16)
#define EPSV  1e-5f

// padded row strides (even for b64 alignment; stride%64 spreads 16 lanes
// across 16 distinct LDS banks -> conflict-free fragment loads)
#define W1S 290             // stage-1 weight row stride (288 used)
#define W2S 218             // stage-2 weight row stride (216 used)
#define US  34              // U channel stride per position (32 used)
#define GS  26              // G channel stride per position (24 used)

// ---- workspace layout (float offsets) ----
#define OFF_W1   0                       // 32*W1S packed stage-1 weights, K = t*32+cu
#define OFF_W2   (32*W1S)                // 16*W2S packed stage-2 weights, K = t*24+c
#define OFF_B1   (OFF_W2 + 16*W2S)       // 32 biases (b5|b10|b15|0)
#define OFF_B2   (OFF_B1 + 32)           // 16 biases (gb|mb)
#define OFF_ST   (OFF_B2 + 16)           // 16 floats: sum[8], sumsq[8]
#define OFF_P5   (OFF_ST + 16)           // 32*8*60*60
#define OFF_P10  (OFF_P5 + 921600)       // 32*8*30*30
#define OFF_P15  (OFF_P10 + 230400)      // 32*8*20*20

// ---- LDS layout (float offsets) ----
#define L_W1 0
#define L_W2 (32*W1S)
#define L_B1 (L_W2 + 16*W2S)
#define L_B2 (L_B1 + 32)
#define L_ST (L_B2 + 16)
#define L_U  (L_ST + 16)                 // [20*20 pos][US]
#define L_G  (L_U + 400*US)              // [18*18 pos][GS]
#define LDS_FLOATS (L_G + 324*GS)

// --------------------------------------------------------------------------
// Kernel 0: pack implicit-GEMM weight matrices (K-order = tap-major,
// channel-minor to match channel-innermost LDS tiles) + biases, zero stats.
// W1 rows 0-7 c5, 8-15 c10, 16-23 c15 over union channels
//   [x(0-7), up5(8-15), up10(16-23), up15(24-31)]; rows 24-31 zero.
// W2 rows 0-7 gw, 8-15 mw over 24 channels.
// --------------------------------------------------------------------------
__global__ void RIA_pack(const float* __restrict__ w5,  const float* __restrict__ b5,
                         const float* __restrict__ w10, const float* __restrict__ b10,
                         const float* __restrict__ w15, const float* __restrict__ b15,
                         const float* __restrict__ gw,  const float* __restrict__ gb,
                         const float* __restrict__ mw,  const float* __restrict__ mb,
                         float* __restrict__ ws) {
  const int tid = threadIdx.x;
  for (int i = tid; i < 32 * W1S; i += 256) {
    int m = i / W1S, kk = i % W1S;
    float v = 0.f;
    if (kk < 288) {
      int t = kk / 32, cu = kk % 32;     // tap-major K order
      if (m < 8) {
        if (cu < 16) v = w5[(m * 16 + cu) * 9 + t];
      } else if (m < 16) {
        int om = m - 8;
        if (cu < 8)                   v = w10[(om * 16 + cu) * 9 + t];
        else if (cu >= 16 && cu < 24) v = w10[(om * 16 + (cu - 8)) * 9 + t];
      } else if (m < 24) {
        int om = m - 16;
        if (cu < 8)        v = w15[(om * 16 + cu) * 9 + t];
        else if (cu >= 24) v = w15[(om * 16 + (cu - 16)) * 9 + t];
      }
    }
    ws[OFF_W1 + i] = v;
  }
  for (int i = tid; i < 16 * W2S; i += 256) {
    int m = i / W2S, kk = i % W2S;
    float v = 0.f;
    if (kk < 216) {
      int t = kk / 24, c = kk % 24;
      v = (m < 8) ? gw[(m * 24 + c) * 9 + t] : mw[((m - 8) * 24 + c) * 9 + t];
    }
    ws[OFF_W2 + i] = v;
  }
  if (tid < 32) {
    float v = 0.f;
    if (tid < 8)       v = b5[tid];
    else if (tid < 16) v = b10[tid - 8];
    else if (tid < 24) v = b15[tid - 16];
    ws[OFF_B1 + tid] = v;
  }
  if (tid < 16) {
    ws[OFF_B2 + tid] = (tid < 8) ? gb[tid] : mb[tid - 8];
    ws[OFF_ST + tid] = 0.f;
  }
}

// --------------------------------------------------------------------------
// Kernel 1: adaptive average pool to (300/k, 300/k).
// --------------------------------------------------------------------------
__global__ void RIA_pool(const float* __restrict__ x, float* __restrict__ p,
                         int k, int P) {
  const int idx = blockIdx.x * 256 + threadIdx.x;
  const int total = BATCH * 8 * P * P;
  if (idx >= total) return;
  int pw = idx % P;
  int t  = idx / P;
  int ph = t % P;  t /= P;
  int c  = t % 8;
  int b  = t / 8;
  const float* src = x + ((size_t)(b * 8 + c) * IMG_H + ph * k) * IMG_W + pw * k;
  float s = 0.f;
  for (int i = 0; i < k; ++i)
    for (int j = 0; j < k; ++j)
      s += src[i * IMG_W + j];
  p[idx] = s / (float)(k * k);
}

// --------------------------------------------------------------------------
// Kernel 2: fused two-stage WMMA conv pipeline per 16x16 output tile.
// --------------------------------------------------------------------------
__global__ __launch_bounds__(256)
void RIA_fused(const float* __restrict__ x, const float* __restrict__ ws,
               float* __restrict__ out, float* __restrict__ gstat) {
  extern __shared__ float smem[];
  float* s_w1 = smem + L_W1;
  float* s_w2 = smem + L_W2;
  float* s_b1 = smem + L_B1;
  float* s_b2 = smem + L_B2;
  float* s_st = smem + L_ST;
  float* s_U  = smem + L_U;    // [pos=20*20][US] channel-innermost
  float* s_G  = smem + L_G;    // [pos=18*18][GS] channel-innermost

  const int tid   = threadIdx.x;
  const int lane  = tid & 31;
  const int wave  = tid >> 5;          // 0..7
  const int lmod  = lane & 15;
  const int lhalf = lane >> 4;         // 0 or 1

  const int b  = blockIdx.z;
  const int h0 = blockIdx.y * 16;
  const int w0 = blockIdx.x * 16;
  const int b8 = b * 8;

  const float* p5  = ws + OFF_P5;
  const float* p10 = ws + OFF_P10;
  const float* p15 = ws + OFF_P15;

  // ---- stage weights + biases into LDS; zero tile stats ----
  for (int i = tid; i < 32 * W1S; i += 256) s_w1[i] = ws[OFF_W1 + i];
  for (int i = tid; i < 16 * W2S; i += 256) s_w2[i] = ws[OFF_W2 + i];
  if (tid < 32) s_b1[tid] = ws[OFF_B1 + tid];
  if (tid < 16) { s_b2[tid] = ws[OFF_B2 + tid]; s_st[tid] = 0.f; }

  // ---- load 20x20 halo of union input (x | up5 | up10 | up15), zero-padded.
  //      Global reads coalesced along w; LDS write scattered channel-inner.
  for (int i = tid; i < 32 * 400; i += 256) {
    int col = i % 20;
    int row = (i / 20) % 20;
    int cu  = i / 400;
    int h = h0 - 2 + row;
    int w = w0 - 2 + col;
    float v = 0.f;
    if (h >= 0 && h < IMG_H && w >= 0 && w < IMG_W) {
      int c = cu & 7;
      if (cu < 8)       v = x  [((size_t)(b8 + c) * IMG_H + h) * IMG_W + w];
      else if (cu < 16) v = p5 [((size_t)(b8 + c) * 60 + h / 5)  * 60 + w / 5];
      else if (cu < 24) v = p10[((size_t)(b8 + c) * 30 + h / 10) * 30 + w / 10];
      else              v = p15[((size_t)(b8 + c) * 20 + h / 15) * 20 + w / 15];
    }
    s_U[(row * 20 + col) * US + cu] = v;
  }
  __syncthreads();

  // ---- stage 1: G[18x18][24] = W1[32][288] x im2col(U)[288][324],
  //      K = t*32 + cu; all fragment loads are immediate-offset b64. ----
  for (int p = wave; p < 42; p += 8) {
    const int mtile = p / 21;
    const int ntile = p % 21;
    const int ncol  = ntile * 16 + lmod;
    const bool nvalid = (ncol < 324);
    const int giy = nvalid ? (ncol / 18) : 0;
    const int gix = nvalid ? (ncol % 18) : 0;
    const float* wrow  = s_w1 + (mtile * 16 + lmod) * W1S + 2 * lhalf;
    const float* ubase = s_U + (giy * 20 + gix) * US + 2 * lhalf;
    v8f acc = {};
#pragma unroll
    for (int t = 0; t < 9; ++t) {
      const int uoff = ((t / 3) * 20 + (t % 3)) * US;
#pragma unroll
      for (int c0 = 0; c0 < 32; c0 += 4) {
        v2f a  = *(const v2f*)(wrow + t * 32 + c0);
        v2f bm = *(const v2f*)(ubase + uoff + c0);
        acc = __builtin_amdgcn_wmma_f32_16x16x4_f32(false, a, false, bm,
                                                    (short)0, acc, false, false);
      }
    }
    const int mbase = mtile * 16 + lhalf * 8;   // {0,8,16,24}
    if (nvalid && mbase < 24) {
      float* gdst = s_G + ncol * GS + mbase;    // 8 consecutive channels
#pragma unroll
      for (int v = 0; v < 8; ++v) gdst[v] = acc[v] + s_b1[mbase + v];
    }
  }
  __syncthreads();

  // ---- stage 2: [gw;mw][16][216] x im2col(G)[216][256]; gate + BN stats ----
  for (int p = wave; p < 16; p += 8) {
    const int ncol = p * 16 + lmod;        // 0..255
    const int oy = ncol >> 4, ox = ncol & 15;
    const float* wrow  = s_w2 + lmod * W2S + 2 * lhalf;
    const float* gbase = s_G + (oy * 18 + ox) * GS + 2 * lhalf;
    v8f acc = {};
#pragma unroll
    for (int t = 0; t < 9; ++t) {
      const int goff = ((t / 3) * 18 + (t % 3)) * GS;
#pragma unroll
      for (int c0 = 0; c0 < 24; c0 += 4) {
        v2f a  = *(const v2f*)(wrow + t * 24 + c0);
        v2f bm = *(const v2f*)(gbase + goff + c0);
        acc = __builtin_amdgcn_wmma_f32_16x16x4_f32(false, a, false, bm,
                                                    (short)0, acc, false, false);
      }
    }
    const int hh = h0 + oy, ww = w0 + ox;
    const bool pvalid = (hh < IMG_H) && (ww < IMG_W);
#pragma unroll
    for (int v = 0; v < 8; ++v) {
      // lanes 16-31 hold mask-conv rows; compute sigmoid there, shuffle down.
      float mval = acc[v] + s_b2[8 + v];
      float sig  = 1.f / (1.f + __expf(-mval));
      float sig_hi = __shfl(sig, lmod + 16, 32);
      if (lhalf == 0 && pvalid) {
        float y = (acc[v] + s_b2[v]) * sig_hi;
        out[((size_t)(b8 + v) * IMG_H + hh) * IMG_W + ww] = y;
        atomicAdd(&s_st[v], y);
        atomicAdd(&s_st[8 + v], y * y);
      }
    }
  }
  __syncthreads();
  if (tid < 16) atomicAdd(&gstat[tid], s_st[tid]);
}

// --------------------------------------------------------------------------
// Kernel 3: batch-norm finalize in place on d_out.
// --------------------------------------------------------------------------
__global__ void RIA_finalize(float* __restrict__ out, const float* __restrict__ stats,
                             const float* __restrict__ gamma, const float* __restrict__ beta) {
  const int idx = blockIdx.x * 256 + threadIdx.x;
  const int total = BATCH * 8 * IMG_H * IMG_W;
  if (idx >= total) return;
  const int c = (idx / (IMG_H * IMG_W)) % 8;
  const float n = (float)(BATCH * IMG_H * IMG_W);
  const float mean = stats[c] / n;
  const float var  = stats[8 + c] / n - mean * mean;
  const float v = out[idx];
  out[idx] = (v - mean) * rsqrtf(var + EPSV) * gamma[c] + beta[c];
}

// --------------------------------------------------------------------------
extern "C" void kernel_launch(void* const* d_in, const int* in_sizes, int n_in,
                              void* d_out, int out_size, void* d_ws, size_t ws_size,
                              hipStream_t stream) {
  const float* x     = (const float*)d_in[0];
  const float* w5    = (const float*)d_in[1];
  const float* b5    = (const float*)d_in[2];
  const float* w10   = (const float*)d_in[3];
  const float* b10   = (const float*)d_in[4];
  const float* w15   = (const float*)d_in[5];
  const float* b15   = (const float*)d_in[6];
  const float* gw    = (const float*)d_in[7];
  const float* gb    = (const float*)d_in[8];
  const float* mw    = (const float*)d_in[9];
  const float* mb    = (const float*)d_in[10];
  const float* gamma = (const float*)d_in[11];
  const float* beta  = (const float*)d_in[12];
  float* ws  = (float*)d_ws;
  float* out = (float*)d_out;

  RIA_pack<<<1, 256, 0, stream>>>(w5, b5, w10, b10, w15, b15, gw, gb, mw, mb, ws);

  RIA_pool<<<(921600 + 255) / 256, 256, 0, stream>>>(x, ws + OFF_P5, 5, 60);
  RIA_pool<<<(230400 + 255) / 256, 256, 0, stream>>>(x, ws + OFF_P10, 10, 30);
  RIA_pool<<<(102400 + 255) / 256, 256, 0, stream>>>(x, ws + OFF_P15, 15, 20);

  const size_t lds_bytes = (size_t)LDS_FLOATS * sizeof(float);
  hipFuncSetAttribute(reinterpret_cast<const void*>(RIA_fused),
                      hipFuncAttributeMaxDynamicSharedMemorySize, (int)lds_bytes);
  dim3 grid(NTX, NTX, BATCH);
  RIA_fused<<<grid, 256, lds_bytes, stream>>>(x, ws, out, ws + OFF_ST);

  const int total = BATCH * 8 * IMG_H * IMG_W;
  RIA_finalize<<<(total + 255) / 256, 256, 0, stream>>>(out, ws + OFF_ST, gamma, beta);
}